// Processor_2619930050630
// MI455X (gfx1250) — compile-verified
//
#include <hip/hip_runtime.h>
#include <hip/hip_bf16.h>

#define H 128
#define N_NODES 50000
#define N_EDGES 640000
#define LN_EPS 1e-5f

typedef __attribute__((ext_vector_type(2))) float v2f;
typedef __attribute__((ext_vector_type(8))) float v8f;
typedef __attribute__((ext_vector_type(4))) int   v4i;

#if __has_builtin(__builtin_amdgcn_global_load_async_to_lds_b128) && \
    __has_builtin(__builtin_amdgcn_s_wait_asynccnt)
#define USE_ASYNC_LDS 1
#else
#define USE_ASYNC_LDS 0
#endif

// -----------------------------------------------------------------------------
// Kernel 0: one-shot repack of W_root / W_nbr into WMMA B-fragment order.
// For n-tile t (0..7), k-step-pair sp (0..15), lane (0..31):
//   pk[((t*16+sp)*32+lane)] = { B(s=2sp).x, B(s=2sp).y, B(s=2sp+1).x, B(s=2sp+1).y }
// where B(s).{x,y} = W[4s + (lane>=16?2:0) + {0,1}][16t + (lane&15)].
// Enables one coalesced global_load_b128 per matrix per 2 K-steps in the GEMM.
// -----------------------------------------------------------------------------
__global__ __launch_bounds__(256) void repack_W(
    const float* __restrict__ Wr, const float* __restrict__ Wn,
    float4* __restrict__ pkR, float4* __restrict__ pkN)
{
    const int tid  = blockIdx.x * 256 + threadIdx.x;   // 0..8191
    const int lane = tid & 31;
    const int sp   = (tid >> 5) & 15;
    const int t    = (tid >> 9) & 7;
    const int mat  = tid >> 12;                        // 0=root, 1=nbr

    const float* W = mat ? Wn : Wr;
    float4*     pk = mat ? pkN : pkR;

    const int n   = t * 16 + (lane & 15);
    const int kA0 = 8 * sp + ((lane >> 4) << 1);
    const int kA1 = kA0 + 4;

    float4 v;
    v.x = W[(size_t)kA0 * H + n];
    v.y = W[(size_t)(kA0 + 1) * H + n];
    v.z = W[(size_t)kA1 * H + n];
    v.w = W[(size_t)(kA1 + 1) * H + n];
    pk[(size_t)(t * 16 + sp) * 32 + lane] = v;
}

// -----------------------------------------------------------------------------
// Kernel 1: fused dual GEMM   msg = X @ W_root + bias ,  xnbr = X @ W_nbr
// Block = 256 threads = 8 waves; 16-row X tile staged in LDS via the async
// VMEM->LDS path (ASYNCcnt), row-padded to 132 floats to avoid bank conflicts.
// Each wave owns one 16-wide N-tile; B fragments come as b128 from packed W.
// -----------------------------------------------------------------------------
__global__ __launch_bounds__(256) void gemm_dual_wmma(
    const float* __restrict__ X,
    const float4* __restrict__ pkR,
    const float4* __restrict__ pkN,
    const float* __restrict__ bias,
    float* __restrict__ msg,
    float* __restrict__ xnbr)
{
    __shared__ float lA[16][132];

    const int tid  = threadIdx.x;
    const int wave = tid >> 5;
    const int lane = tid & 31;
    const int m0   = blockIdx.x * 16;

    // Stage 16x128 A tile: 512 float4's across 256 threads.
    for (int i = tid; i < 16 * 32; i += 256) {
        const int r  = i >> 5;
        const int c4 = i & 31;
#if USE_ASYNC_LDS
        __builtin_amdgcn_global_load_async_to_lds_b128(
            (__attribute__((address_space(1))) v4i*)((v4i*)(X + (size_t)(m0 + r) * H + c4 * 4)),
            (__attribute__((address_space(3))) v4i*)((v4i*)&lA[r][c4 * 4]),
            0, 0);
#else
        float4 v = ((const float4*)(X + (size_t)(m0 + r) * H))[c4];
        *(float4*)&lA[r][c4 * 4] = v;
#endif
    }
#if USE_ASYNC_LDS
    __builtin_amdgcn_s_wait_asynccnt(0);
#endif
    __syncthreads();

    const int l  = lane & 15;          // row (A) / col (B,D) within tile
    const int kk = (lane >> 4) * 2;    // lanes 0-15: K=0,1 ; lanes 16-31: K=2,3

    const float4* pR = pkR + (size_t)(wave * 16) * 32 + lane;
    const float4* pN = pkN + (size_t)(wave * 16) * 32 + lane;

    v8f accR = {};
    v8f accN = {};

    #pragma unroll 4
    for (int sp = 0; sp < 16; ++sp) {      // each sp covers K-steps 2sp, 2sp+1
        const float4 br4 = pR[sp * 32];
        const float4 bn4 = pN[sp * 32];
        const int k0 = sp * 8;

        v2f a0, a1;
        a0.x = lA[l][k0 + kk];
        a0.y = lA[l][k0 + kk + 1];
        a1.x = lA[l][k0 + 4 + kk];
        a1.y = lA[l][k0 + 4 + kk + 1];

        v2f br0 = {br4.x, br4.y}, br1 = {br4.z, br4.w};
        v2f bn0 = {bn4.x, bn4.y}, bn1 = {bn4.z, bn4.w};

        accR = __builtin_amdgcn_wmma_f32_16x16x4_f32(
            false, a0, false, br0, (short)0, accR, false, false);
        accN = __builtin_amdgcn_wmma_f32_16x16x4_f32(
            false, a0, false, bn0, (short)0, accN, false, false);
        accR = __builtin_amdgcn_wmma_f32_16x16x4_f32(
            false, a1, false, br1, (short)0, accR, false, false);
        accN = __builtin_amdgcn_wmma_f32_16x16x4_f32(
            false, a1, false, bn1, (short)0, accN, false, false);
    }

    // C/D layout: VGPR i -> lanes 0-15: (M=i, N=lane), lanes 16-31: (M=8+i, N=lane-16)
    const int rbase = m0 + ((lane >> 4) << 3);
    const int col   = wave * 16 + l;
    const float bv  = bias[col];
    #pragma unroll
    for (int i = 0; i < 8; ++i) {
        const size_t off = (size_t)(rbase + i) * H + col;
        msg[off]  = accR[i] + bv;
        xnbr[off] = accN[i];
    }
}

// -----------------------------------------------------------------------------
// Kernel 2: edge scatter-add.  One wave per edge: gather 512B row of xnbr[src]
// as float4/lane, atomically accumulate into msg[dst] (global_atomic_add_f32).
// -----------------------------------------------------------------------------
__global__ __launch_bounds__(256) void edge_scatter(
    const int* __restrict__ edge_index,   // [2, N_EDGES]
    const float* __restrict__ xnbr,
    float* __restrict__ msg)
{
    const int wid  = (blockIdx.x * blockDim.x + threadIdx.x) >> 5;
    const int lane = threadIdx.x & 31;
    if (wid >= N_EDGES) return;

    const int src = edge_index[wid];
    const int dst = edge_index[N_EDGES + wid];

    const float4 v = ((const float4*)(xnbr + (size_t)src * H))[lane];
    float* p = msg + (size_t)dst * H + lane * 4;
    unsafeAtomicAdd(p + 0, v.x);
    unsafeAtomicAdd(p + 1, v.y);
    unsafeAtomicAdd(p + 2, v.z);
    unsafeAtomicAdd(p + 3, v.w);
}

// -----------------------------------------------------------------------------
// Kernel 3: GELU(exact) + residual + LayerNorm.  One wave per node row,
// 4 floats per lane, wave shuffle reduction for mean/var.
// -----------------------------------------------------------------------------
__device__ __forceinline__ float gelu_exact(float v) {
    return 0.5f * v * (1.0f + erff(v * 0.70710678118654752440f));
}

__global__ __launch_bounds__(256) void finalize_ln(
    const float* __restrict__ msg,
    const float* __restrict__ X,
    const float* __restrict__ gamma,
    const float* __restrict__ beta,
    float* __restrict__ out)
{
    const int row  = (blockIdx.x * blockDim.x + threadIdx.x) >> 5;
    const int lane = threadIdx.x & 31;
    if (row >= N_NODES) return;

    const float4 m4 = ((const float4*)(msg + (size_t)row * H))[lane];
    const float4 x4 = ((const float4*)(X + (size_t)row * H))[lane];

    float h0 = gelu_exact(m4.x) + x4.x;
    float h1 = gelu_exact(m4.y) + x4.y;
    float h2 = gelu_exact(m4.z) + x4.z;
    float h3 = gelu_exact(m4.w) + x4.w;

    float s  = h0 + h1 + h2 + h3;
    float ss = h0 * h0 + h1 * h1 + h2 * h2 + h3 * h3;
    #pragma unroll
    for (int off = 16; off > 0; off >>= 1) {
        s  += __shfl_xor(s, off, 32);
        ss += __shfl_xor(ss, off, 32);
    }
    const float mean = s * (1.0f / H);
    const float var  = ss * (1.0f / H) - mean * mean;
    const float inv  = rsqrtf(var + LN_EPS);

    const float4 g4 = ((const float4*)gamma)[lane];
    const float4 b4 = ((const float4*)beta)[lane];

    float4 o;
    o.x = (h0 - mean) * inv * g4.x + b4.x;
    o.y = (h1 - mean) * inv * g4.y + b4.y;
    o.z = (h2 - mean) * inv * g4.z + b4.z;
    o.w = (h3 - mean) * inv * g4.w + b4.w;
    ((float4*)(out + (size_t)row * H))[lane] = o;
}

// -----------------------------------------------------------------------------
// Launch. d_in order: x_hidden, edge_index, W_root, W_nbr, bias, ln_gamma, ln_beta
// ws layout: [0) xnbr 25.6MB | msg 25.6MB | pkWr 64KB | pkWn 64KB
// -----------------------------------------------------------------------------
extern "C" void kernel_launch(void* const* d_in, const int* in_sizes, int n_in,
                              void* d_out, int out_size, void* d_ws, size_t ws_size,
                              hipStream_t stream) {
    const float* X     = (const float*)d_in[0];
    const int*   ei    = (const int*)d_in[1];
    const float* Wr    = (const float*)d_in[2];
    const float* Wn    = (const float*)d_in[3];
    const float* bias  = (const float*)d_in[4];
    const float* gamma = (const float*)d_in[5];
    const float* beta  = (const float*)d_in[6];
    float* out = (float*)d_out;

    float*  xnbr = (float*)d_ws;
    float*  msg  = xnbr + (size_t)N_NODES * H;
    float4* pkR  = (float4*)(msg + (size_t)N_NODES * H);
    float4* pkN  = pkR + (size_t)8 * 16 * 32;   // 4096 float4 per matrix

    // 0) Repack W into WMMA B-fragment order (2*8*16*32 = 8192 threads).
    repack_W<<<32, 256, 0, stream>>>(Wr, Wn, pkR, pkN);

    // 1) Dual GEMM via f32 WMMA: 50000/16 = 3125 row tiles.
    gemm_dual_wmma<<<N_NODES / 16, 256, 0, stream>>>(X, pkR, pkN, bias, msg, xnbr);

    // 2) Edge scatter: 1 wave/edge, 8 waves/block.
    edge_scatter<<<(N_EDGES + 7) / 8, 256, 0, stream>>>(ei, xnbr, msg);

    // 3) GELU + residual + LayerNorm: 1 wave/row, 8 rows/block.
    finalize_ln<<<(N_NODES + 7) / 8, 256, 0, stream>>>(msg, X, gamma, beta, out);
}